// DigitCapsules_36661840839125
// MI455X (gfx1250) — compile-verified
//
#include <hip/hip_runtime.h>
#include <math.h>

// CapsNet dynamic routing, fused for MI455X (gfx1250, wave32).
// u_hat is recomputed each routing iteration with V_WMMA_F32_16X16X4_F32
// instead of being materialized (256MB fp32) -> HBM traffic collapses from
// ~1.8GB to ~24MB; x (8MB) + W (16MB) stay L2-resident across the 3 passes.
// b_ij is never stored: logit(t) = sum_o u_hat * Vsum where Vsum = sum of
// previous v iterates (128KB), so the 16MB b_ij tensor disappears too.

typedef float v2f __attribute__((ext_vector_type(2)));
typedef float v8f __attribute__((ext_vector_type(8)));

#define B_      128
#define IC_     2048
#define NC_     16
#define OD_     16
#define ID_     8
#define BT_     16                 // batch tile (WMMA N)
#define NCHUNK_MAX_ 64             // preferred IC split (512 waves in flight)

// ---------------------------------------------------------------------------
// Kernel A: one wave (32 threads) per (batch-tile, IC-chunk).
// For each i in chunk: u_hat tile per capsule c is D[o=M, b=N] computed via
// two v_wmma_f32_16x16x4_f32 (K=8 = 2x4). Routing logits, softmax over c,
// and weighted accumulation into s all stay in registers.
// Accumulator layout (32-bit C/D 16x16): lane<16 -> M=r, N=lane;
// lane>=16 -> M=r+8, N=lane-16. Here M=o, N=b.
// ---------------------------------------------------------------------------
__global__ __launch_bounds__(32)
void caps_route_partial(const float* __restrict__ x,
                        const float* __restrict__ W,
                        const float* __restrict__ Vsum,
                        float* __restrict__ s_partial,
                        int ic_per_chunk)
{
    const int btile = blockIdx.x;        // 0..7
    const int chunk = blockIdx.y;        // 0..nchunk-1
    const int lane  = threadIdx.x & 31;
    const int half  = lane >> 4;         // 0 or 1
    const int l16   = lane & 15;
    const int b0    = btile * BT_;
    const int bme   = b0 + l16;          // this lane's batch row (N index)

    // Vsum tiles in D layout: Vs[c][r] = Vsum[bme, c, o = r + half*8]
    v8f Vs[NC_];
#pragma unroll
    for (int c = 0; c < NC_; ++c) {
        const float* p = Vsum + ((bme * NC_) + c) * OD_ + half * 8;
        v8f t;
#pragma unroll
        for (int r = 0; r < 8; ++r) t[r] = p[r];
        Vs[c] = t;
    }

    // s accumulator tiles, same layout as D
    v8f sacc[NC_];
#pragma unroll
    for (int c = 0; c < NC_; ++c) sacc[c] = (v8f)0.0f;

    const int ibase = chunk * ic_per_chunk;
    for (int ii = 0; ii < ic_per_chunk; ++ii) {
        const int i = ibase + ii;

        // B operand = x^T (4x16): lane<16 -> K=v, N=lane; lane>=16 -> K=v+2.
        // So per lane: b = l16, d = half*2 + v  (k-step 1 adds 4).
        const float* xp = x + ((size_t)bme * IC_ + i) * ID_ + half * 2;
        v2f xb0; xb0[0] = xp[0]; xb0[1] = xp[1];
        v2f xb1; xb1[0] = xp[4]; xb1[1] = xp[5];

        // Prefetch next i's W block into cache (global_prefetch_b8)
        if (ii + 1 < ic_per_chunk) {
            __builtin_prefetch(W + ((size_t)(i + 1) * NC_) * OD_ * ID_, 0, 3);
        }

        v8f D[NC_];
        float lg[NC_];
#pragma unroll
        for (int c = 0; c < NC_; ++c) {
            // A operand = W[i,c] (16x4): M=o=l16, K = half*2 + v (+4 on kstep1)
            const float* wp = W + (((size_t)i * NC_ + c) * OD_ + l16) * ID_ + half * 2;
            v2f wa0; wa0[0] = wp[0]; wa0[1] = wp[1];
            v2f wa1; wa1[0] = wp[4]; wa1[1] = wp[5];

            v8f d = (v8f)0.0f;
            d = __builtin_amdgcn_wmma_f32_16x16x4_f32(false, wa0, false, xb0,
                                                      (short)0, d, false, false);
            d = __builtin_amdgcn_wmma_f32_16x16x4_f32(false, wa1, false, xb1,
                                                      (short)0, d, false, false);
            D[c] = d;

            // logit[b,c] = sum_o u_hat * Vsum : reduce over r, then across halves
            float t = 0.0f;
#pragma unroll
            for (int r = 0; r < 8; ++r) t += d[r] * Vs[c][r];
            t += __shfl_xor(t, 16, 32);
            lg[c] = t;
        }

        // softmax over c (16 scalars per lane; both halves hold same b -> consistent)
        float m = lg[0];
#pragma unroll
        for (int c = 1; c < NC_; ++c) m = fmaxf(m, lg[c]);
        float ssum = 0.0f;
#pragma unroll
        for (int c = 0; c < NC_; ++c) { lg[c] = __expf(lg[c] - m); ssum += lg[c]; }
        const float inv = 1.0f / ssum;

#pragma unroll
        for (int c = 0; c < NC_; ++c) {
            const float cij = lg[c] * inv;
#pragma unroll
            for (int r = 0; r < 8; ++r) sacc[c][r] += cij * D[c][r];
        }
    }

    // write partial s for this chunk: s_partial[chunk][b][c][o]
#pragma unroll
    for (int c = 0; c < NC_; ++c) {
        float* p = s_partial + ((((size_t)chunk * B_) + bme) * NC_ + c) * OD_ + half * 8;
#pragma unroll
        for (int r = 0; r < 8; ++r) p[r] = sacc[c][r];
    }
}

// ---------------------------------------------------------------------------
// Kernel B: reduce the nchunk partials, squash, Vsum += v, optionally emit out.
// One thread per (b, c).
// ---------------------------------------------------------------------------
__global__ void caps_squash_update(const float* __restrict__ s_partial,
                                   float* __restrict__ Vsum,
                                   float* __restrict__ out,
                                   int nchunk, int write_out)
{
    const int idx = blockIdx.x * blockDim.x + threadIdx.x; // over B*NC
    if (idx >= B_ * NC_) return;

    float sv[OD_];
#pragma unroll
    for (int o = 0; o < OD_; ++o) sv[o] = 0.0f;
    for (int k = 0; k < nchunk; ++k) {
        const float* p = s_partial + ((size_t)k * B_ * NC_ + idx) * OD_;
#pragma unroll
        for (int o = 0; o < OD_; ++o) sv[o] += p[o];
    }

    float sq = 0.0f;
#pragma unroll
    for (int o = 0; o < OD_; ++o) sq += sv[o] * sv[o];
    const float scale = (sq / (1.0f + sq)) / sqrtf(sq + 1e-8f);

    float* vp = Vsum + (size_t)idx * OD_;
#pragma unroll
    for (int o = 0; o < OD_; ++o) {
        const float v = scale * sv[o];
        vp[o] += v;
        if (write_out) out[(size_t)idx * OD_ + o] = v;
    }
}

__global__ void caps_zero(float* __restrict__ p, int n)
{
    const int idx = blockIdx.x * blockDim.x + threadIdx.x;
    if (idx < n) p[idx] = 0.0f;
}

// ---------------------------------------------------------------------------
extern "C" void kernel_launch(void* const* d_in, const int* in_sizes, int n_in,
                              void* d_out, int out_size, void* d_ws, size_t ws_size,
                              hipStream_t stream)
{
    const float* x = (const float*)d_in[0];   // [128, 2048, 8]
    const float* W = (const float*)d_in[1];   // [1, 2048, 16, 16, 8]
    float* out = (float*)d_out;               // [128, 16, 16]

    const size_t tile_floats = (size_t)B_ * NC_ * OD_;   // 32768 (128KB)

    // Pick the largest power-of-two chunk count whose partial buffers fit ws.
    int nchunk = NCHUNK_MAX_;
    while (nchunk > 1 && (size_t)(nchunk + 1) * tile_floats * sizeof(float) > ws_size)
        nchunk >>= 1;
    const int ic_per_chunk = IC_ / nchunk;

    float* Vsum      = (float*)d_ws;              // tile_floats
    float* s_partial = Vsum + tile_floats;        // nchunk * tile_floats

    const int vn = (int)tile_floats;
    caps_zero<<<(vn + 255) / 256, 256, 0, stream>>>(Vsum, vn);

    dim3 gridA(B_ / BT_, nchunk);   // up to 8 x 64 = 512 waves
    for (int t = 0; t < 3; ++t) {
        caps_route_partial<<<gridA, 32, 0, stream>>>(x, W, Vsum, s_partial,
                                                     ic_per_chunk);
        caps_squash_update<<<(B_ * NC_ + 255) / 256, 256, 0, stream>>>(
            s_partial, Vsum, out, nchunk, (t == 2) ? 1 : 0);
    }
}